// TensorTrainComplexOutput_55886114455739
// MI455X (gfx1250) — compile-verified
//
#include <hip/hip_runtime.h>
#include <math.h>

typedef float v2f __attribute__((ext_vector_type(2)));
typedef float v4f __attribute__((ext_vector_type(4)));
typedef float v8f __attribute__((ext_vector_type(8)));

#define IN_F   128
#define OUT_F  32000
#define BATCH  2048
#define TT_R   8

// ---------------------------------------------------------------------------
// Packed fragment layout for V_WMMA_F32_16X16X4_F32 (A and B share the map):
//   lanes 0-15  : index = lane,    K in {0,1}  (VGPR0=K0, VGPR1=K1)
//   lanes 16-31 : index = lane-16, K in {2,3}  (VGPR0=K2, VGPR1=K3)
// Two consecutive k-steps are interleaved into one float4 per lane so each
// fragment-pair is a single coalesced b128.
// ---------------------------------------------------------------------------
__device__ __forceinline__ int pack_off(int rowTile, int row15, int k) {
    const int kstep = k >> 2;            // 0..31
    const int kk    = k & 3;             // 0..3 within a WMMA K=4 step
    const int sp    = kstep >> 1;        // k-pair 0..15
    const int sl    = kstep & 1;         // which step of the pair
    const int lane  = ((kk >> 1) << 4) | row15;
    return (((rowTile * 16 + sp) * 32 + lane) << 2) + (sl << 1) + (kk & 1);
}

// ---------------------------------------------------------------------------
// Kernel 1: reconstruct Wr / Wi from TT factors straight into fragment layout.
// W[o,i] = sum_r g1[0,a,b,r] * g2[r,c,d,r] * g3[r,e,f,0]
// ---------------------------------------------------------------------------
__global__ __launch_bounds__(256)
void tt_pack_w(const float* __restrict__ g1r, const float* __restrict__ g2r,
               const float* __restrict__ g3r, const float* __restrict__ g1i,
               const float* __restrict__ g2i, const float* __restrict__ g3i,
               float* __restrict__ wr, float* __restrict__ wi) {
    const int idx = blockIdx.x * 256 + threadIdx.x;       // < OUT_F * IN_F
    const int n = idx >> 7;
    const int k = idx & 127;
    const int a = n / 1600, c = (n / 40) % 40, e = n % 40;
    const int b = k >> 5,   d = (k >> 3) & 3, f = k & 7;
    float sr = 0.f, si = 0.f;
#pragma unroll
    for (int r = 0; r < TT_R; ++r) {
        const int i1 = (a * 4 + b) * 8 + r;               // g1 (1,20,4,8)
        const int i2 = ((r * 40 + c) * 4 + d) * 8 + r;    // g2 diag (8,40,4,8)
        const int i3 = (r * 40 + e) * 8 + f;              // g3 (8,40,8,1)
        sr += g1r[i1] * g2r[i2] * g3r[i3];
        si += g1i[i1] * g2i[i2] * g3i[i3];
    }
    const int off = pack_off(n >> 4, n & 15, k);
    wr[off] = sr;
    wi[off] = si;
}

// ---------------------------------------------------------------------------
// Kernel 2: pack x_real / x_imag into A-fragment layout.
// ---------------------------------------------------------------------------
__global__ __launch_bounds__(256)
void tt_pack_x(const float* __restrict__ xr, const float* __restrict__ xi,
               float* __restrict__ xrp, float* __restrict__ xip) {
    const int idx = blockIdx.x * 256 + threadIdx.x;       // < BATCH * IN_F
    const int m = idx >> 7;
    const int k = idx & 127;
    const int off = pack_off(m >> 4, m & 15, k);
    xrp[off] = xr[idx];
    xip[off] = xi[idx];
}

// ---------------------------------------------------------------------------
// CDNA5 async global->LDS staging (ASYNCcnt-tracked, no VGPR staging).
// GV addressing: global_load_async_to_lds_b128 vdst_lds, vaddr64, off
// LDS address = low 32 bits of the generic shared pointer (ISA: LDS aperture
// maps addr[31:0] to the LDS offset).
// ---------------------------------------------------------------------------
__device__ __forceinline__ void async_b128_to_lds(v4f* lds_dst, const v4f* gsrc) {
    const unsigned           lds = (unsigned)(size_t)lds_dst;
    const unsigned long long ga  = (unsigned long long)(size_t)gsrc;
    asm volatile("global_load_async_to_lds_b128 %0, %1, off"
                 :: "v"(lds), "v"(ga) : "memory");
}

__device__ __forceinline__ void wait_asynccnt0() {
#if __has_builtin(__builtin_amdgcn_s_wait_asynccnt)
    __builtin_amdgcn_s_wait_asynccnt(0);
#else
    asm volatile("s_wait_asynccnt 0x0" ::: "memory");
#endif
}

// ---------------------------------------------------------------------------
// Kernel 3: fused complex GEMM + log-magnitude epilogue.
// Block tile: 128 M x 64 N. The 64 KB W strip (Wr+Wi, fragment-packed and
// contiguous) is async-loaded into LDS once and shared by all 8 waves.
// Each wave: 16 M x 64 N -> 512 v_wmma_f32_16x16x4_f32.
//   real += xr*Wr + (-xi)*Wi     imag += xi*Wr + xr*Wi
// ---------------------------------------------------------------------------
#define WMMA_F32(A, B, C) \
    __builtin_amdgcn_wmma_f32_16x16x4_f32(false, (A), false, (B), (short)0, (C), false, false)

__global__ __launch_bounds__(256)
void tt_gemm(const float* __restrict__ wr, const float* __restrict__ wi,
             const float* __restrict__ xrp, const float* __restrict__ xip,
             float* __restrict__ out) {
    __shared__ v4f sB[4096];                              // [0:2048)=Wr, [2048:4096)=Wi

    const int lane  = threadIdx.x & 31;
    const int wave  = threadIdx.x >> 5;
    const int nblk  = blockIdx.x % (OUT_F / 64);          // 0..499 : 64-col strip
    const int mblk  = blockIdx.x / (OUT_F / 64);          // 0..15  : 128-row group
    const int mtile = mblk * 8 + wave;                    // 0..127

    const v4f* __restrict__ xr4 = (const v4f*)xrp;
    const v4f* __restrict__ xi4 = (const v4f*)xip;
    const v4f* __restrict__ wr4 = (const v4f*)wr;
    const v4f* __restrict__ wi4 = (const v4f*)wi;

    // --- async fill of the 64 KB B strip: 2048 v4f per component -----------
    {
        const int strip = nblk * 2048;                    // v4f index of strip base
#pragma unroll
        for (int c = 0; c < 8; ++c) {
            const int j = threadIdx.x + 256 * c;          // 0..2047
            async_b128_to_lds(&sB[j],        wr4 + strip + j);
            async_b128_to_lds(&sB[2048 + j], wi4 + strip + j);
        }
        wait_asynccnt0();
        __syncthreads();
    }

    v8f accR[4] = {};
    v8f accI[4] = {};

#pragma unroll 2
    for (int sp = 0; sp < 16; ++sp) {                     // K pairs (8 k's each)
        const int abase = (mtile * 16 + sp) * 32 + lane;
        const v4f ar = xr4[abase];
        const v4f ai = xi4[abase];
        const v2f ar0 = {ar[0], ar[1]}, ar1 = {ar[2], ar[3]};
        const v2f ai0 = {ai[0], ai[1]}, ai1 = {ai[2], ai[3]};
        const v2f nai0 = {-ai[0], -ai[1]}, nai1 = {-ai[2], -ai[3]};
#pragma unroll
        for (int t = 0; t < 4; ++t) {
            const int bidx = (t * 16 + sp) * 32 + lane;
            const v4f br = sB[bidx];
            const v4f bi = sB[2048 + bidx];
            const v2f br0 = {br[0], br[1]}, br1 = {br[2], br[3]};
            const v2f bi0 = {bi[0], bi[1]}, bi1 = {bi[2], bi[3]};
            // k-step 2*sp
            accR[t] = WMMA_F32(ar0,  br0, accR[t]);
            accR[t] = WMMA_F32(nai0, bi0, accR[t]);
            accI[t] = WMMA_F32(ai0,  br0, accI[t]);
            accI[t] = WMMA_F32(ar0,  bi0, accI[t]);
            // k-step 2*sp+1
            accR[t] = WMMA_F32(ar1,  br1, accR[t]);
            accR[t] = WMMA_F32(nai1, bi1, accR[t]);
            accI[t] = WMMA_F32(ai1,  br1, accI[t]);
            accI[t] = WMMA_F32(ar1,  bi1, accI[t]);
        }
    }

    // Epilogue: logits = log(re^2 + im^2 + 1e-10), fast hw log (v_log_f32).
    // C/D layout: VGPR v -> row m0 + v (+8 for lanes 16-31), col n0 + lane%16
    const int row0 = mtile * 16 + ((lane >> 4) << 3);
    const int col0 = nblk * 64 + (lane & 15);
#pragma unroll
    for (int t = 0; t < 4; ++t) {
#pragma unroll
        for (int v = 0; v < 8; ++v) {
            const float re = accR[t][v];
            const float im = accI[t][v];
            const float val = __logf(re * re + im * im + 1e-10f);
            out[(size_t)(row0 + v) * OUT_F + (col0 + t * 16)] = val;
        }
    }
}

// ---------------------------------------------------------------------------
extern "C" void kernel_launch(void* const* d_in, const int* in_sizes, int n_in,
                              void* d_out, int out_size, void* d_ws, size_t ws_size,
                              hipStream_t stream) {
    (void)in_sizes; (void)n_in; (void)out_size; (void)ws_size;
    const float* x_real = (const float*)d_in[0];
    const float* x_imag = (const float*)d_in[1];
    const float* g1r    = (const float*)d_in[2];
    const float* g2r    = (const float*)d_in[3];
    const float* g3r    = (const float*)d_in[4];
    const float* g1i    = (const float*)d_in[5];
    const float* g2i    = (const float*)d_in[6];
    const float* g3i    = (const float*)d_in[7];
    float* out = (float*)d_out;

    float* ws  = (float*)d_ws;
    float* wrp = ws;                                   // 32000*128 = 4,096,000 f
    float* wip = ws + (size_t)OUT_F * IN_F;            // 4,096,000 f
    float* xrp = ws + (size_t)2 * OUT_F * IN_F;        // 262,144 f
    float* xip = xrp + (size_t)BATCH * IN_F;           // 262,144 f  (~34.9 MB)

    tt_pack_w<<<(OUT_F * IN_F) / 256, 256, 0, stream>>>(g1r, g2r, g3r, g1i, g2i, g3i, wrp, wip);
    tt_pack_x<<<(BATCH * IN_F) / 256, 256, 0, stream>>>(x_real, x_imag, xrp, xip);

    // (2048/128) row-groups * (32000/64) col-strips = 8000 blocks of 8 waves
    tt_gemm<<<(BATCH / 128) * (OUT_F / 64), 256, 0, stream>>>(wrp, wip, xrp, xip, out);
}